// GATEncoder_19121194401844
// MI455X (gfx1250) — compile-verified
//
#include <hip/hip_runtime.h>

#define IN_C  128
#define OUT_C 128
#define NEG_SLOPE 0.2f
#define EPS_DEN 1e-16f

typedef __attribute__((ext_vector_type(2))) float v2f;
typedef __attribute__((ext_vector_type(8))) float v8f;

// -------- segment-max via sign-aware integer atomics (sound for mixed signs) --------
__device__ __forceinline__ void atomicMaxF32(float* addr, float val) {
    if (val >= 0.0f) atomicMax((int*)addr, __float_as_int(val));
    else             atomicMin((unsigned int*)addr, __float_as_uint(val));
}

// ---------------- Kernel 1: h = x @ W via V_WMMA_F32_16X16X4_F32 --------------------
// Block = 128 threads (4 waves). Block computes a 16-row stripe of h (16 x 128).
// Wave w computes columns [w*32, w*32+32) as two 16x16 WMMA tiles, K-loop step 4.
__global__ __launch_bounds__(128) void gat_gemm(const float* __restrict__ x,
                                                const float* __restrict__ W,
                                                float* __restrict__ h,
                                                int N) {
    const int lane  = threadIdx.x & 31;
    const int wave  = threadIdx.x >> 5;        // 0..3
    const int l16   = lane & 15;
    const int khalf = lane >> 4;               // 0: K={0,1}, 1: K={2,3}
    const int rowbase = blockIdx.x * 16;
    if (rowbase + 16 > N) return;              // N divisible by 16 in this problem
    const int ncol0 = wave * 32;

    const float* __restrict__ xrow = x + (size_t)(rowbase + l16) * IN_C + khalf * 2;

    v8f acc0 = {};
    v8f acc1 = {};

    #pragma unroll 4
    for (int k0 = 0; k0 < IN_C; k0 += 4) {
        // A fragment: 16x4 fp32.  VGPR0 = K=(khalf*2), VGPR1 = K=(khalf*2+1)
        v2f a;
        a.x = xrow[k0 + 0];
        a.y = xrow[k0 + 1];
        // B fragments: 4x16 fp32, row-striped across lanes
        const int kb = k0 + khalf * 2;
        v2f b0, b1;
        b0.x = W[(size_t)(kb + 0) * OUT_C + ncol0 + l16];
        b0.y = W[(size_t)(kb + 1) * OUT_C + ncol0 + l16];
        b1.x = W[(size_t)(kb + 0) * OUT_C + ncol0 + 16 + l16];
        b1.y = W[(size_t)(kb + 1) * OUT_C + ncol0 + 16 + l16];
        acc0 = __builtin_amdgcn_wmma_f32_16x16x4_f32(false, a, false, b0,
                                                     (short)0, acc0, false, false);
        acc1 = __builtin_amdgcn_wmma_f32_16x16x4_f32(false, a, false, b1,
                                                     (short)0, acc1, false, false);
    }

    // C/D layout: VGPR r holds M=r (lanes 0-15) and M=r+8 (lanes 16-31), N=lane%16
    #pragma unroll
    for (int r = 0; r < 8; ++r) {
        const int row = rowbase + r + khalf * 8;
        h[(size_t)row * OUT_C + ncol0 + l16]      = acc0[r];
        h[(size_t)row * OUT_C + ncol0 + 16 + l16] = acc1[r];
    }
}

// ------------- Kernel 2: alpha_s[i] = h[i,:].a_src ; alpha_d[i] = h[i,:].a_dst ------
__global__ __launch_bounds__(256) void gat_alpha(const float* __restrict__ h,
                                                 const float* __restrict__ a_src,
                                                 const float* __restrict__ a_dst,
                                                 float* __restrict__ as_,
                                                 float* __restrict__ ad_,
                                                 int N) {
    const int row = blockIdx.x * 8 + (threadIdx.x >> 5);
    if (row >= N) return;
    const int lane = threadIdx.x & 31;
    float4 hv = ((const float4*)(h + (size_t)row * OUT_C))[lane];
    float4 s4 = ((const float4*)a_src)[lane];
    float4 d4 = ((const float4*)a_dst)[lane];
    float s = hv.x * s4.x + hv.y * s4.y + hv.z * s4.z + hv.w * s4.w;
    float d = hv.x * d4.x + hv.y * d4.y + hv.z * d4.z + hv.w * d4.w;
    #pragma unroll
    for (int off = 16; off > 0; off >>= 1) {
        s += __shfl_xor(s, off, 32);
        d += __shfl_xor(d, off, 32);
    }
    if (lane == 0) { as_[row] = s; ad_[row] = d; }
}

// ------------- Kernel 3: zero out, init m=-inf, denom=0 ----------------------------
__global__ void gat_init(float* __restrict__ out, float* __restrict__ m,
                         float* __restrict__ denom, int N) {
    const int idx = blockIdx.x * blockDim.x + threadIdx.x;
    if (idx < N * OUT_C) out[idx] = 0.0f;
    if (idx < N) { m[idx] = -__builtin_inff(); denom[idx] = 0.0f; }
}

// ------------- Kernel 4: e = leakyrelu(as[src]+ad[dst]); m[dst]=max ----------------
__global__ void gat_edge_max(const int* __restrict__ ei, int E, int N,
                             const float* __restrict__ as_,
                             const float* __restrict__ ad_,
                             float* __restrict__ ebuf,
                             float* __restrict__ m) {
    const int idx = blockIdx.x * blockDim.x + threadIdx.x;
    const int total = E + N;
    if (idx >= total) return;
    int src, dst;
    if (idx < E) { src = ei[idx]; dst = ei[E + idx]; }
    else         { src = idx - E; dst = src; }          // self loops
    float e = as_[src] + ad_[dst];
    e = (e > 0.0f) ? e : NEG_SLOPE * e;
    ebuf[idx] = e;
    atomicMaxF32(m + dst, e);
}

// ------------- Kernel 5: ex = exp(e - m[dst]); denom[dst] += ex --------------------
__global__ void gat_edge_sum(const int* __restrict__ ei, int E, int N,
                             const float* __restrict__ m,
                             float* __restrict__ ebuf,
                             float* __restrict__ denom) {
    const int idx = blockIdx.x * blockDim.x + threadIdx.x;
    const int total = E + N;
    if (idx >= total) return;
    const int dst = (idx < E) ? ei[E + idx] : (idx - E);
    float ex = expf(ebuf[idx] - m[dst]);
    ebuf[idx] = ex;
    atomicAdd(denom + dst, ex);
}

// ------------- Kernel 6: out[dst,:] += h[src,:] * (ex / (denom[dst]+eps)) ----------
// One wave32 per edge; lane handles 4 consecutive features (coalesced 512B/wave).
__global__ __launch_bounds__(256) void gat_edge_agg(const int* __restrict__ ei,
                                                    int E, int N,
                                                    const float* __restrict__ ebuf,
                                                    const float* __restrict__ denom,
                                                    const float* __restrict__ h,
                                                    float* __restrict__ out) {
    const int wid = blockIdx.x * 8 + (threadIdx.x >> 5);
    const int total = E + N;
    if (wid >= total) return;
    const int lane = threadIdx.x & 31;
    int src, dst;
    if (wid < E) { src = ei[wid]; dst = ei[E + wid]; }
    else         { src = wid - E; dst = src; }
    const float alpha = ebuf[wid] / (denom[dst] + EPS_DEN);
    float4 hv = ((const float4*)(h + (size_t)src * OUT_C))[lane];
    float* op = out + (size_t)dst * OUT_C + lane * 4;
    atomicAdd(op + 0, hv.x * alpha);
    atomicAdd(op + 1, hv.y * alpha);
    atomicAdd(op + 2, hv.z * alpha);
    atomicAdd(op + 3, hv.w * alpha);
}

// ------------- Kernel 7: out = relu(out + bias) ------------------------------------
__global__ void gat_final(float* __restrict__ out, const float* __restrict__ bias,
                          int N) {
    const int idx = blockIdx.x * blockDim.x + threadIdx.x;
    if (idx >= N * OUT_C) return;
    float v = out[idx] + bias[idx & (OUT_C - 1)];
    out[idx] = (v > 0.0f) ? v : 0.0f;
}

extern "C" void kernel_launch(void* const* d_in, const int* in_sizes, int n_in,
                              void* d_out, int out_size, void* d_ws, size_t ws_size,
                              hipStream_t stream) {
    const float* x     = (const float*)d_in[0];
    const float* W     = (const float*)d_in[1];
    const float* a_src = (const float*)d_in[2];
    const float* a_dst = (const float*)d_in[3];
    const float* bias  = (const float*)d_in[4];
    const int*   ei    = (const int*)d_in[5];

    const int N = in_sizes[0] / IN_C;      // 50000
    const int E = in_sizes[5] / 2;         // 1600000
    const int total = E + N;

    // Workspace layout (floats)
    float* ws    = (float*)d_ws;
    float* h     = ws;                      // N*128
    float* as_   = h + (size_t)N * OUT_C;   // N
    float* ad_   = as_ + N;                 // N
    float* m     = ad_ + N;                 // N
    float* denom = m + N;                   // N
    float* ebuf  = denom + N;               // E+N
    float* out   = (float*)d_out;

    // 1. GEMM (WMMA fp32)
    gat_gemm<<<N / 16, 128, 0, stream>>>(x, W, h, N);
    // 2. per-node attention logits
    gat_alpha<<<(N + 7) / 8, 256, 0, stream>>>(h, a_src, a_dst, as_, ad_, N);
    // 3. init accumulators
    gat_init<<<(N * OUT_C + 255) / 256, 256, 0, stream>>>(out, m, denom, N);
    // 4. segment max
    gat_edge_max<<<(total + 255) / 256, 256, 0, stream>>>(ei, E, N, as_, ad_, ebuf, m);
    // 5. exp + segment sum
    gat_edge_sum<<<(total + 255) / 256, 256, 0, stream>>>(ei, E, N, m, ebuf, denom);
    // 6. weighted aggregation (wave per edge)
    gat_edge_agg<<<(total + 7) / 8, 256, 0, stream>>>(ei, E, N, ebuf, denom, h, out);
    // 7. bias + relu
    gat_final<<<(N * OUT_C + 255) / 256, 256, 0, stream>>>(out, bias, N);
}